// DAGLayer_77412490543905
// MI455X (gfx1250) — compile-verified
//
#include <hip/hip_runtime.h>
#include <math.h>

typedef __attribute__((ext_vector_type(2))) float v2f;
typedef __attribute__((ext_vector_type(8))) float v8f;

#define NEGV (-1e30f)

// ---------- helpers ----------
__device__ __forceinline__ unsigned fenc(float f) {
    unsigned u = __float_as_uint(f);
    return (u >> 31) ? ~u : (u | 0x80000000u);
}
__device__ __forceinline__ float fdec(unsigned u) {
    u = (u >> 31) ? (u & 0x7FFFFFFFu) : ~u;
    return __uint_as_float(u);
}
__device__ __forceinline__ float sigm(float x) { return 1.0f / (1.0f + expf(-x)); }

// ---------- WMMA fp32 strip GEMM: C[M, NT*16] = A[M,K] @ W[NT*16, K]^T + bias ----------
// One wave computes a 16-row x (NT*16)-col strip: A fragment loaded once per k-step and
// reused across NT accumulators (NT WMMAs per A load). M mult of 16, K mult of 4.
template <int NT>
__global__ void wmma_gemm_strip(const float* __restrict__ A, const float* __restrict__ W,
                                const float* __restrict__ bias, float* __restrict__ C,
                                int M, int K) {
    const int Nout = NT * 16;
    int wave = blockIdx.x * (blockDim.x >> 5) + (threadIdx.x >> 5);
    if (wave >= (M >> 4)) return;              // wave-uniform: EXEC stays all-ones
    int lane = threadIdx.x & 31;
    int half = lane >> 4;                       // 0: lanes 0-15, 1: lanes 16-31
    int lm   = lane & 15;

    const float* Arow = A + (size_t)(wave * 16 + lm) * K;  // A row M = lm

    v8f acc[NT];
#pragma unroll
    for (int t = 0; t < NT; ++t) acc[t] = (v8f){};

    for (int kb = 0; kb < K; kb += 4) {
        int k0 = kb + 2 * half;                 // VGPR v holds k = kb + 2*half + v
        v2f a;
        a.x = Arow[k0]; a.y = Arow[k0 + 1];
#pragma unroll
        for (int t = 0; t < NT; ++t) {
            const float* Wrow = W + (size_t)(t * 16 + lm) * K;  // B col N = lm
            v2f b;
            b.x = Wrow[k0]; b.y = Wrow[k0 + 1];
            acc[t] = __builtin_amdgcn_wmma_f32_16x16x4_f32(false, a, false, b,
                                                           (short)0, acc[t], false, false);
        }
    }
#pragma unroll
    for (int t = 0; t < NT; ++t) {
        int col = t * 16 + lm;
        float bv = bias[col];
        float* Crow = C + (size_t)(wave * 16 + half * 8) * Nout + col;  // row = 16w + 8*half + r
#pragma unroll
        for (int r = 0; r < 8; ++r)
            Crow[(size_t)r * Nout] = acc[t][r] + bv;
    }
}

// ---------- elementwise / edge kernels ----------
__global__ void set_u32(unsigned* p, unsigned v, int n) {
    int i = blockIdx.x * blockDim.x + threadIdx.x;
    if (i < n) p[i] = v;
}
__global__ void mark_zero_at(const int* __restrict__ idxs, unsigned* __restrict__ flags, int n) {
    int i = blockIdx.x * blockDim.x + threadIdx.x;
    if (i < n) flags[idxs[i]] = 0u;
}

// h0 = root ? (1-z)*tanh(gi_n + r*bhh_n) : 0   (GRU with zero hidden)
__global__ void h0_kernel(const float* __restrict__ gi, const float* __restrict__ bhh,
                          const unsigned* __restrict__ root, float* __restrict__ h, int N) {
    int i = blockIdx.x * blockDim.x + threadIdx.x;
    if (i >= N * 64) return;
    int n = i >> 6, d = i & 63;
    const float* g = gi + (size_t)n * 192;
    float r  = sigm(g[d]       + bhh[d]);
    float z  = sigm(g[64 + d]  + bhh[64 + d]);
    float nn = tanhf(g[128 + d] + r * bhh[128 + d]);
    h[i] = root[n] ? (1.0f - z) * nn : 0.0f;
}

__global__ void level_init(float* __restrict__ msg, unsigned* __restrict__ dstmax,
                           float* __restrict__ den, int N) {
    int i = blockIdx.x * blockDim.x + threadIdx.x;
    if (i < N * 64) msg[i] = 0.0f;
    if (i < N) { dstmax[i] = fenc(NEGV); den[i] = 0.0f; }
}

// One wave per edge: logit_e = ba + Wa[:64].q + Wa[64:].(h[src]+eemb), coalesced loads,
// wave32 shfl reduction; lane 0 stores logit + ordered-uint atomic segment-max.
__global__ void edge_logit_max(const int* __restrict__ src, const int* __restrict__ dst,
                               const unsigned char* __restrict__ mask,
                               const float* __restrict__ x, const float* __restrict__ h,
                               const float* __restrict__ eemb, const float* __restrict__ Wa,
                               const float* __restrict__ ba, float* __restrict__ logit,
                               unsigned* __restrict__ dstmax, int E) {
    int e = blockIdx.x * (blockDim.x >> 5) + (threadIdx.x >> 5);
    if (e >= E) return;                        // wave-uniform
    if (!mask[e]) return;                      // wave-uniform
    int lane = threadIdx.x & 31;
    int s = src[e], dd = dst[e];
    const float* q  = x    + (size_t)dd * 64;
    const float* hs = h    + (size_t)s  * 64;
    const float* ee = eemb + (size_t)e  * 64;
    int d0 = lane, d1 = lane + 32;
    float acc = Wa[d0]      * q[d0] + Wa[64 + d0] * (hs[d0] + ee[d0])
              + Wa[d1]      * q[d1] + Wa[64 + d1] * (hs[d1] + ee[d1]);
#pragma unroll
    for (int off = 16; off > 0; off >>= 1)
        acc += __shfl_down(acc, off, 32);
    if (lane == 0) {
        float lg = acc + ba[0];
        logit[e] = lg;
        atomicMax(dstmax + dd, fenc(lg));
    }
}

__global__ void edge_exp_sum(const int* __restrict__ dst, const unsigned char* __restrict__ mask,
                             const float* __restrict__ logit, const unsigned* __restrict__ dstmax,
                             float* __restrict__ exbuf, float* __restrict__ den, int E) {
    int e = blockIdx.x * blockDim.x + threadIdx.x;
    if (e >= E || !mask[e]) return;
    int dd = dst[e];
    float ex = expf(logit[e] - fdec(dstmax[dd]));
    exbuf[e] = ex;
    atomicAdd(den + dd, ex);
}

__global__ void edge_scatter_msg(const int* __restrict__ src, const int* __restrict__ dst,
                                 const unsigned char* __restrict__ mask,
                                 const float* __restrict__ exbuf, const float* __restrict__ den,
                                 const float* __restrict__ h, float* __restrict__ msg, int E) {
    long long i = (long long)blockIdx.x * blockDim.x + threadIdx.x;
    if (i >= (long long)E * 64) return;
    int e = (int)(i >> 6), d = (int)(i & 63);
    if (!mask[e]) return;
    int s = src[e], dd = dst[e];
    float alpha = exbuf[e] / (den[dd] + 1e-16f);
    atomicAdd(msg + (size_t)dd * 64 + d, alpha * h[(size_t)s * 64 + d]);
}

// h = nm ? (1-z)*tanh(gi_n + r*gh_n) + z*msg : h    where nm = (den > 0)
__global__ void gru_select(const float* __restrict__ gi, const float* __restrict__ gh,
                           const float* __restrict__ msg, const float* __restrict__ den,
                           float* __restrict__ h, int N) {
    int i = blockIdx.x * blockDim.x + threadIdx.x;
    if (i >= N * 64) return;
    int n = i >> 6, d = i & 63;
    if (!(den[n] > 0.0f)) return;
    const float* g = gi + (size_t)n * 192;
    const float* G = gh + (size_t)n * 192;
    float r  = sigm(g[d]       + G[d]);
    float z  = sigm(g[64 + d]  + G[64 + d]);
    float nn = tanhf(g[128 + d] + r * G[128 + d]);
    h[i] = (1.0f - z) * nn + z * msg[i];
}

// ---------- leaf index compaction (reproduces jnp.nonzero(size=NL) order, 0-padded) ----------
__global__ void leaf_count(const unsigned* __restrict__ flags, unsigned* __restrict__ blockcnt, int N) {
    int i = blockIdx.x * 256 + threadIdx.x;
    unsigned f = (i < N) ? flags[i] : 0u;
    unsigned long long bal = __ballot(f != 0u);
    __shared__ unsigned wc[8];
    int lane = threadIdx.x & 31, wid = threadIdx.x >> 5;
    if (lane == 0) wc[wid] = (unsigned)__popcll(bal);
    __syncthreads();
    if (threadIdx.x == 0) {
        unsigned t = 0;
        for (int w = 0; w < 8; ++w) t += wc[w];
        blockcnt[blockIdx.x] = t;
    }
}
__global__ void scan_single(const unsigned* __restrict__ blockcnt, unsigned* __restrict__ blockoff, int nb) {
    if (blockIdx.x == 0 && threadIdx.x == 0) {
        unsigned run = 0;
        for (int b = 0; b < nb; ++b) { blockoff[b] = run; run += blockcnt[b]; }
    }
}
__global__ void leaf_write(const unsigned* __restrict__ flags, const unsigned* __restrict__ blockoff,
                           int* __restrict__ idx, int N, int NLmax) {
    int i = blockIdx.x * 256 + threadIdx.x;
    unsigned f = (i < N) ? flags[i] : 0u;
    unsigned long long bal = __ballot(f != 0u);
    __shared__ unsigned wc[8], woff[8];
    int lane = threadIdx.x & 31, wid = threadIdx.x >> 5;
    if (lane == 0) wc[wid] = (unsigned)__popcll(bal);
    __syncthreads();
    if (threadIdx.x == 0) {
        unsigned run = 0;
        for (int w = 0; w < 8; ++w) { woff[w] = run; run += wc[w]; }
    }
    __syncthreads();
    if (f) {
        unsigned rank = blockoff[blockIdx.x] + woff[wid] +
                        (unsigned)__popcll(bal & ((1ull << lane) - 1ull));
        if ((int)rank < NLmax) idx[rank] = i;
    }
}

__global__ void leaf_gather(const int* __restrict__ idx1, const int* __restrict__ idx2,
                            const float* __restrict__ o1, const float* __restrict__ o2,
                            float* __restrict__ cat, int NL) {
    int i = blockIdx.x * blockDim.x + threadIdx.x;
    if (i >= NL * 128) return;
    int r = i >> 7, c = i & 127;
    cat[i] = (c < 64) ? o1[(size_t)idx1[r] * 64 + c]
                      : o2[(size_t)idx2[r] * 64 + (c - 64)];
}
__global__ void leaf_scatter(const int* __restrict__ idx1, const int* __restrict__ idx2,
                             const float* __restrict__ feat, float* __restrict__ o1,
                             float* __restrict__ o2, int NL) {
    int i = blockIdx.x * blockDim.x + threadIdx.x;
    if (i >= NL * 64) return;
    int r = i >> 6, d = i & 63;
    o1[(size_t)idx1[r] * 64 + d] = feat[(size_t)r * 128 + d];
    o2[(size_t)idx2[r] * 64 + d] = feat[(size_t)r * 128 + 64 + d];
}

// ---------- host orchestration ----------
extern "C" void kernel_launch(void* const* d_in, const int* in_sizes, int n_in,
                              void* d_out, int out_size, void* d_ws, size_t ws_size,
                              hipStream_t stream) {
    const int D = 64;
    const int N = in_sizes[0] / D;       // 100000
    const int E = in_sizes[1] / 2;       // 400000
    const int L = in_sizes[3] / E;       // 4
    const int NL = 10000;                // n_leaf (fixed scalar in harness setup)

    const float*         x1   = (const float*)d_in[0];
    const int*           ei1  = (const int*)d_in[1];
    const float*         ea1  = (const float*)d_in[2];
    const unsigned char* em1  = (const unsigned char*)d_in[3];
    const float*         x2   = (const float*)d_in[4];
    const int*           ei2  = (const int*)d_in[5];
    const float*         ea2  = (const float*)d_in[6];
    const unsigned char* em2  = (const unsigned char*)d_in[7];
    const float* We  = (const float*)d_in[8];
    const float* be  = (const float*)d_in[9];
    const float* Wa  = (const float*)d_in[10];
    const float* ba  = (const float*)d_in[11];
    const float* Wih = (const float*)d_in[12];
    const float* Whh = (const float*)d_in[13];
    const float* bih = (const float*)d_in[14];
    const float* bhh = (const float*)d_in[15];
    const float* Wc  = (const float*)d_in[16];
    const float* bc  = (const float*)d_in[17];

    float* o1 = (float*)d_out;
    float* o2 = o1 + (size_t)N * D;

    // ---- workspace carve ----
    char* p = (char*)d_ws;
    auto alloc = [&](size_t bytes) -> void* {
        void* r = (void*)p;
        p += (bytes + 255) & ~((size_t)255);
        return r;
    };
    float*    eemb    = (float*)   alloc((size_t)E * D * 4);
    float*    gi      = (float*)   alloc((size_t)N * 192 * 4);
    float*    gh      = (float*)   alloc((size_t)N * 192 * 4);
    float*    msg     = (float*)   alloc((size_t)N * D * 4);
    float*    logit   = (float*)   alloc((size_t)E * 4);
    float*    exbuf   = (float*)   alloc((size_t)E * 4);
    float*    den     = (float*)   alloc((size_t)N * 4);
    unsigned* dstmax  = (unsigned*)alloc((size_t)N * 4);
    unsigned* flags   = (unsigned*)alloc((size_t)N * 4);
    int*      idx1    = (int*)     alloc((size_t)NL * 4);
    int*      idx2    = (int*)     alloc((size_t)NL * 4);
    const int nb = (N + 255) / 256;
    unsigned* blockcnt = (unsigned*)alloc((size_t)nb * 4);
    unsigned* blockoff = (unsigned*)alloc((size_t)nb * 4);
    float*    catbuf   = (float*)  alloc((size_t)NL * 128 * 4);
    float*    feat     = (float*)  alloc((size_t)NL * 128 * 4);
    (void)ws_size; (void)n_in; (void)out_size;

    auto cdiv = [](long long a, long long b) -> int { return (int)((a + b - 1) / b); };
    // Strip GEMM: one wave per 16 rows, Nout = NT*16 fixed per instantiation.
    auto gemm = [&](const float* A, const float* W, const float* bias, float* C,
                    int M, int K, int Nout) {
        int blocks = cdiv(M / 16, 8);           // 8 waves per 256-thread block
        switch (Nout) {
            case 64:  wmma_gemm_strip<4> <<<blocks, 256, 0, stream>>>(A, W, bias, C, M, K); break;
            case 128: wmma_gemm_strip<8> <<<blocks, 256, 0, stream>>>(A, W, bias, C, M, K); break;
            case 192: wmma_gemm_strip<12><<<blocks, 256, 0, stream>>>(A, W, bias, C, M, K); break;
        }
    };

    auto run_graph = [&](const float* x, const int* ei, const float* ea,
                         const unsigned char* em, float* h) {
        const int* src = ei;
        const int* dst = ei + E;
        // root mask: nodes with no incoming edge
        set_u32<<<cdiv(N, 256), 256, 0, stream>>>(flags, 1u, N);
        mark_zero_at<<<cdiv(E, 256), 256, 0, stream>>>(dst, flags, E);
        // eemb = edge_attr @ We^T + be        [E,64]
        gemm(ea, We, be, eemb, E, 64, 64);
        // gi = x @ Wih^T + bih                [N,192]  (level-invariant, hoisted)
        gemm(x, Wih, bih, gi, N, 64, 192);
        // h0
        h0_kernel<<<cdiv((long long)N * 64, 256), 256, 0, stream>>>(gi, bhh, flags, h, N);
        for (int l = 0; l < L; ++l) {
            const unsigned char* m = em + (size_t)l * E;
            level_init<<<cdiv((long long)N * 64, 256), 256, 0, stream>>>(msg, dstmax, den, N);
            edge_logit_max<<<cdiv(E, 8), 256, 0, stream>>>(src, dst, m, x, h, eemb,
                                                           Wa, ba, logit, dstmax, E);
            edge_exp_sum<<<cdiv(E, 256), 256, 0, stream>>>(dst, m, logit, dstmax, exbuf, den, E);
            edge_scatter_msg<<<cdiv((long long)E * 64, 256), 256, 0, stream>>>(
                src, dst, m, exbuf, den, h, msg, E);
            // gh = msg @ Whh^T + bhh          [N,192]
            gemm(msg, Whh, bhh, gh, N, 64, 192);
            gru_select<<<cdiv((long long)N * 64, 256), 256, 0, stream>>>(gi, gh, msg, den, h, N);
        }
    };

    auto find_leaves = [&](const int* src, int* idx) {
        set_u32<<<cdiv(N, 256), 256, 0, stream>>>(flags, 1u, N);
        mark_zero_at<<<cdiv(E, 256), 256, 0, stream>>>(src, flags, E);
        set_u32<<<cdiv(NL, 256), 256, 0, stream>>>((unsigned*)idx, 0u, NL); // nonzero 0-pad
        leaf_count<<<nb, 256, 0, stream>>>(flags, blockcnt, N);
        scan_single<<<1, 32, 0, stream>>>(blockcnt, blockoff, nb);
        leaf_write<<<nb, 256, 0, stream>>>(flags, blockoff, idx, N, NL);
    };

    run_graph(x1, ei1, ea1, em1, o1);
    run_graph(x2, ei2, ea2, em2, o2);

    find_leaves(ei1, idx1);
    find_leaves(ei2, idx2);

    leaf_gather<<<cdiv((long long)NL * 128, 256), 256, 0, stream>>>(idx1, idx2, o1, o2, catbuf, NL);
    // feat = cat @ Wc^T + bc                 [NL,128]
    gemm(catbuf, Wc, bc, feat, NL, 128, 128);
    leaf_scatter<<<cdiv((long long)NL * 64, 256), 256, 0, stream>>>(idx1, idx2, feat, o1, o2, NL);
}